// Fusion_strage_72928544686148
// MI455X (gfx1250) — compile-verified
//
#include <hip/hip_runtime.h>
#include <hip/hip_bf16.h>

typedef __attribute__((ext_vector_type(2))) float v2f;
typedef __attribute__((ext_vector_type(8))) float v8f;

#define EPS 1e-10f

__device__ __forceinline__ v8f wmma_f32x4(v2f a, v2f b, v8f c) {
  // V_WMMA_F32_16X16X4_F32 : D = A(16x4) * B(4x16) + C(16x16), fp32
  return __builtin_amdgcn_wmma_f32_16x16x4_f32(
      /*neg_a=*/false, a, /*neg_b=*/false, b,
      /*c_mod=*/(short)0, c, /*reuse_a=*/false, /*reuse_b=*/false);
}

struct EPtrs { const float* p[8]; };

// ---------------------------------------------------------------------------
// Kernel 0a: repack conv1 weights [co][ci][kh][kw] -> wpack[tap][ci][co]
// so A-fragment loads in conv1 are lane-coalesced (co contiguous).
// ---------------------------------------------------------------------------
__global__ __launch_bounds__(256) void k_repack_w(
    const float* __restrict__ w1, float* __restrict__ wpack)
{
  int i = blockIdx.x * 256 + threadIdx.x;    // 9*128*64 = 73728
  int tap = i >> 13;                         // [0,9)
  int ci  = (i >> 6) & 127;
  int co  = i & 63;
  wpack[i] = w1[co * 1152 + ci * 9 + tap];
}

// ---------------------------------------------------------------------------
// Kernel 0b: zero-pad fc into fcp[b][ci][130][130] (1-px halo) so conv1's
// K-loop has NO boundary predication: every B load is unconditional.
// grid = (67, 512): y = b*128+ci, x covers 130*130=16900 elems.
// ---------------------------------------------------------------------------
__global__ __launch_bounds__(256) void k_pad_fc(
    const float* __restrict__ fc, float* __restrict__ fcp)
{
  int idx = blockIdx.x * 256 + threadIdx.x;
  if (idx >= 16900) return;
  int bc  = blockIdx.y;                      // b*128 + ci
  int row = idx / 130;
  int col = idx - row * 130;
  float v = 0.0f;
  if (row >= 1 && row <= 128 && col >= 1 && col <= 128)
    v = fc[(size_t)bc * 16384 + (row - 1) * 128 + (col - 1)];
  fcp[(size_t)bc * 16900 + idx] = v;
}

// ---------------------------------------------------------------------------
// Kernel 1: conv1 (128->64, 3x3, SAME) as 9 shifted GEMMs with f32 WMMA on
// the padded input. Per tap: C(16co x 64px) += Wtap(16co x 128ci) * shift.
// 4 accumulators (4 pixel groups) share each A fragment. Zero branches,
// zero masks in the K loop.
// grid = 4 b * 4 co-tiles * 256 px64-tiles = 4096 blocks of 1 wave.
// ---------------------------------------------------------------------------
__global__ __launch_bounds__(32) void k_conv1_wmma(
    const float* __restrict__ fcp, const float* __restrict__ wpack,
    const float* __restrict__ b1, float* __restrict__ cat)
{
  int blk = blockIdx.x;
  int bb  = blk >> 10;             // batch
  int rem = blk & 1023;
  int cot = rem >> 8;              // out-channel tile (16 ch)
  int pt  = rem & 255;             // 64-pixel tile
  int p0 = pt * 64;
  int y  = p0 >> 7;
  int x0 = p0 & 127;

  int lane = threadIdx.x;
  int half = (lane >> 4) << 1;     // 0 or 2 (K sub-offset per ISA layout)
  int l15  = lane & 15;
  int co_a = cot * 16 + l15;       // A row (out channel) for this lane

  const float* fcpb = fcp + (size_t)bb * 128 * 16900;

  v8f acc0 = {}, acc1 = {}, acc2 = {}, acc3 = {};

#pragma unroll
  for (int kh = 0; kh < 3; ++kh) {
#pragma unroll
    for (int kw = 0; kw < 3; ++kw) {
      int tap = kh * 3 + kw;
      const float* wt = wpack + (size_t)tap * 8192;            // [ci][co]
      // padded coords: prow = y+kh in [0,129], pcol = x0+kw+... in [0,129]
      const float* rb = fcpb + (size_t)(y + kh) * 130 + (x0 + kw) + l15;

      for (int ci = 0; ci < 128; ci += 4) {
        int k0 = ci + half;
        v2f a;
        a.x = wt[k0 * 64 + co_a];                // lane-coalesced
        a.y = wt[(k0 + 1) * 64 + co_a];
        const float* r0 = rb + (size_t)k0 * 16900;
        const float* r1 = r0 + 16900;
        v2f b0, b1v, b2, b3;
        b0.x  = r0[0];  b0.y  = r1[0];
        b1v.x = r0[16]; b1v.y = r1[16];
        b2.x  = r0[32]; b2.y  = r1[32];
        b3.x  = r0[48]; b3.y  = r1[48];
        acc0 = wmma_f32x4(a, b0,  acc0);
        acc1 = wmma_f32x4(a, b1v, acc1);
        acc2 = wmma_f32x4(a, b2,  acc2);
        acc3 = wmma_f32x4(a, b3,  acc3);
      }
    }
  }

  int hi = (lane >> 4) * 8;        // C rows: lanes 16-31 hold M+8
#pragma unroll
  for (int r = 0; r < 8; ++r) {
    int co = cot * 16 + r + hi;
    float bias = b1[co];
    size_t base = ((size_t)bb * 64 + co) * 16384 + y * 128 + x0 + l15;
    float w;
    w = acc0[r] + bias; cat[base +  0] = (w > 0.f) ? w : 0.01f * w;
    w = acc1[r] + bias; cat[base + 16] = (w > 0.f) ? w : 0.01f * w;
    w = acc2[r] + bias; cat[base + 32] = (w > 0.f) ? w : 0.01f * w;
    w = acc3[r] + bias; cat[base + 48] = (w > 0.f) ? w : 0.01f * w;
  }
}

// ---------------------------------------------------------------------------
// Kernel 2: conv2 (64->1, 3x3, SAME), direct. One thread per output pixel.
// ---------------------------------------------------------------------------
__global__ __launch_bounds__(256) void k_conv2(
    const float* __restrict__ cat, const float* __restrict__ w2,
    const float* __restrict__ b2, float* __restrict__ am)
{
  int i = blockIdx.x * 256 + threadIdx.x;   // 65536 total
  int b = i >> 14;
  int p = i & 16383;
  int y = p >> 7, x = p & 127;
  float s = b2[0];
  for (int ci = 0; ci < 64; ++ci) {
    const float* base = cat + ((size_t)b * 64 + ci) * 16384;
    const float* wci  = w2 + ci * 9;
#pragma unroll
    for (int kh = 0; kh < 3; ++kh) {
      int yy = y + kh - 1;
      if ((unsigned)yy >= 128u) continue;
#pragma unroll
      for (int kw = 0; kw < 3; ++kw) {
        int xx = x + kw - 1;
        if ((unsigned)xx >= 128u) continue;
        s += base[yy * 128 + xx] * wci[kh * 3 + kw];
      }
    }
  }
  am[i] = s;
}

// ---------------------------------------------------------------------------
// Kernel 3: global min/max of activation_maps (65536 elems), one block.
// ---------------------------------------------------------------------------
__global__ __launch_bounds__(1024) void k_nam_minmax(
    const float* __restrict__ am, float* __restrict__ st)
{
  int tid = threadIdx.x;
  float mn = 3.4e38f, mx = -3.4e38f;
  for (int i = tid; i < 65536; i += 1024) {
    float v = am[i];
    mn = fminf(mn, v);
    mx = fmaxf(mx, v);
  }
  __shared__ float smn[1024], smx[1024];
  smn[tid] = mn; smx[tid] = mx;
  __syncthreads();
  for (int o = 512; o > 0; o >>= 1) {
    if (tid < o) {
      smn[tid] = fminf(smn[tid], smn[tid + o]);
      smx[tid] = fmaxf(smx[tid], smx[tid + o]);
    }
    __syncthreads();
  }
  if (tid == 0) { st[0] = smn[0]; st[1] = smx[0]; }
}

// ---------------------------------------------------------------------------
// Kernel 4a: per-E partial sum/min/max (for mean + norm_1 scalars).
// ---------------------------------------------------------------------------
__global__ __launch_bounds__(256) void k_estats_part(
    EPtrs eps, float* __restrict__ psum, float* __restrict__ pmin,
    float* __restrict__ pmax)
{
  int e   = blockIdx.x >> 8;
  int blk = blockIdx.x & 255;
  int tid = threadIdx.x;
  const float* f = eps.p[e];
  float s = 0.0f, mn = 3.4e38f, mx = -3.4e38f;
  for (int i = blk * 256 + tid; i < 4194304; i += 65536) {
    float v = f[i];
    s += v; mn = fminf(mn, v); mx = fmaxf(mx, v);
  }
  __shared__ float ss[256], smn[256], smx[256];
  ss[tid] = s; smn[tid] = mn; smx[tid] = mx;
  __syncthreads();
  for (int o = 128; o > 0; o >>= 1) {
    if (tid < o) {
      ss[tid] += ss[tid + o];
      smn[tid] = fminf(smn[tid], smn[tid + o]);
      smx[tid] = fmaxf(smx[tid], smx[tid + o]);
    }
    __syncthreads();
  }
  if (tid == 0) {
    psum[e * 256 + blk] = ss[0];
    pmin[e * 256 + blk] = smn[0];
    pmax[e * 256 + blk] = smx[0];
  }
}

// Kernel 4b: finalize per-E scalars: mean, mn_t, mx_t
__global__ __launch_bounds__(256) void k_estats_final(
    const float* __restrict__ psum, const float* __restrict__ pmin,
    const float* __restrict__ pmax, float* __restrict__ estats)
{
  int e = blockIdx.x, tid = threadIdx.x;
  __shared__ float ss[256], smn[256], smx[256];
  ss[tid]  = psum[e * 256 + tid];
  smn[tid] = pmin[e * 256 + tid];
  smx[tid] = pmax[e * 256 + tid];
  __syncthreads();
  for (int o = 128; o > 0; o >>= 1) {
    if (tid < o) {
      ss[tid] += ss[tid + o];
      smn[tid] = fminf(smn[tid], smn[tid + o]);
      smx[tid] = fmaxf(smx[tid], smx[tid + o]);
    }
    __syncthreads();
  }
  if (tid == 0) {
    float mean = ss[0] * (1.0f / 4194304.0f);
    estats[e * 4 + 0] = mean;
    estats[e * 4 + 1] = fmaxf(smn[0] - mean, EPS);  // mn_t
    estats[e * 4 + 2] = fmaxf(smx[0] - mean, EPS);  // mx_t
  }
}

// ---------------------------------------------------------------------------
// Kernel 5a: Gram scores[e][b] = fr * fr^T (64x64, K=16384), WMMA.
// One wave computes a 16c x 64d strip (4 accumulators share each A load).
// K split 16 ways into a deterministic partials buffer.
// grid = 8e * 4b * 4ctile * 16kchunk = 2048 blocks of 1 wave.
// ---------------------------------------------------------------------------
__global__ __launch_bounds__(32) void k_gram_wmma(
    EPtrs eps, float* __restrict__ spart)
{
  int blk    = blockIdx.x;
  int kchunk = blk & 15;
  int t      = blk >> 4;           // [0,128): e*16 + b*4 + ctile
  int e     = t >> 4;
  int rem   = t & 15;
  int bb    = rem >> 2;
  int ctile = rem & 3;
  int c0 = ctile * 16;

  const float* f = eps.p[e] + (size_t)bb * 64 * 16384;
  int lane = threadIdx.x;
  int half = (lane >> 4) << 1;
  int l15  = lane & 15;
  const float* arow = f + (size_t)(c0 + l15) * 16384;
  const float* br0  = f + (size_t)(0  + l15) * 16384;   // d-tile 0
  const float* br1  = f + (size_t)(16 + l15) * 16384;   // d-tile 1
  const float* br2  = f + (size_t)(32 + l15) * 16384;   // d-tile 2
  const float* br3  = f + (size_t)(48 + l15) * 16384;   // d-tile 3

  v8f a0 = {}, a1 = {}, a2 = {}, a3 = {};
  int kbeg = kchunk * 1024, kend = kbeg + 1024;
  for (int k = kbeg; k < kend; k += 4) {
    int kh = k + half;
    v2f a, b0, b1v, b2, b3;
    a.x = arow[kh];  a.y = arow[kh + 1];
    b0.x  = br0[kh]; b0.y  = br0[kh + 1];
    b1v.x = br1[kh]; b1v.y = br1[kh + 1];
    b2.x  = br2[kh]; b2.y  = br2[kh + 1];
    b3.x  = br3[kh]; b3.y  = br3[kh + 1];
    a0 = wmma_f32x4(a, b0,  a0);
    a1 = wmma_f32x4(a, b1v, a1);
    a2 = wmma_f32x4(a, b2,  a2);
    a3 = wmma_f32x4(a, b3,  a3);
  }

  // spart layout: [kchunk][tile(512)][256], tile = e*64 + b*16 + ctile*4 + dt
  int tbase = e * 64 + bb * 16 + ctile * 4;
  int hi = (lane >> 4) * 8;
  float* dst = spart + (size_t)kchunk * 512 * 256;
#pragma unroll
  for (int r = 0; r < 8; ++r) {
    int ridx = (r + hi) * 16 + l15;
    dst[(size_t)(tbase + 0) * 256 + ridx] = a0[r];
    dst[(size_t)(tbase + 1) * 256 + ridx] = a1[r];
    dst[(size_t)(tbase + 2) * 256 + ridx] = a2[r];
    dst[(size_t)(tbase + 3) * 256 + ridx] = a3[r];
  }
}

// Kernel 5b: deterministic reduction of the 16 K-chunks into scores.
__global__ __launch_bounds__(256) void k_gram_reduce(
    const float* __restrict__ spart, float* __restrict__ scores)
{
  int t   = blockIdx.x;            // tile id [0,512)
  int idx = threadIdx.x;           // element within 16x16 tile
  float s = 0.0f;
#pragma unroll
  for (int k = 0; k < 16; ++k)
    s += spart[((size_t)k * 512 + t) * 256 + idx];
  int e    = t >> 6;
  int rem  = t & 63;
  int bb   = rem >> 4;
  int tile = rem & 15;
  int c = (tile >> 2) * 16 + (idx >> 4);
  int d = (tile & 3) * 16 + (idx & 15);
  scores[(((size_t)e * 4 + bb) * 64 + c) * 64 + d] = s;
}

// ---------------------------------------------------------------------------
// Kernel 6: softmax over axis=1 (rows c) per column (b,d); in-place.
// ---------------------------------------------------------------------------
__global__ __launch_bounds__(256) void k_softmax(float* __restrict__ scores)
{
  int e  = blockIdx.x;
  int t  = threadIdx.x;
  int bb = t >> 6, d = t & 63;
  float* s = scores + (((size_t)e * 4 + bb) * 64) * 64 + d;  // stride 64 over c
  float mx = -3.4e38f;
  for (int c = 0; c < 64; ++c) mx = fmaxf(mx, s[c * 64]);
  float sum = 0.0f;
  for (int c = 0; c < 64; ++c) sum += expf(s[c * 64] - mx);
  float inv = 1.0f / sum;
  for (int c = 0; c < 64; ++c) s[c * 64] = expf(s[c * 64] - mx) * inv;
}

// ---------------------------------------------------------------------------
// Kernel 7: fw = W * fr (64x16384, K=64) via WMMA + fused intr() epilogue
//   out = norm_1(max(f - mean, EPS)) + fw + f   written into M1/M2.
// One wave computes 16c x 64px (4 accumulators share each W load).
// grid = 8e * 4b * 4ctile * 256 px64-tiles = 32768 blocks of 1 wave.
// ---------------------------------------------------------------------------
__global__ __launch_bounds__(32) void k_fw_wmma(
    EPtrs eps, const float* __restrict__ W, const float* __restrict__ estats,
    float* __restrict__ out)
{
  int blk  = blockIdx.x;
  int e    = blk >> 12;
  int rem  = blk & 4095;
  int bb   = rem >> 10;
  int rem2 = rem & 1023;
  int c0 = (rem2 >> 8) * 16;
  int p0 = (rem2 & 255) * 64;

  int lane = threadIdx.x;
  int half = (lane >> 4) << 1;
  int l15  = lane & 15;

  const float* f  = eps.p[e] + (size_t)bb * 64 * 16384;
  const float* Wb = W + ((size_t)e * 4 + bb) * 4096;         // [c][d]
  const float* arow = Wb + (size_t)(c0 + l15) * 64;

  v8f a0 = {}, a1 = {}, a2 = {}, a3 = {};
  for (int k = 0; k < 64; k += 4) {
    int kh = k + half;
    v2f a, b0, b1v, b2, b3;
    a.x = arow[kh];  a.y = arow[kh + 1];
    const float* r0 = f + (size_t)kh * 16384 + p0 + l15;
    const float* r1 = f + (size_t)(kh + 1) * 16384 + p0 + l15;
    b0.x  = r0[0];  b0.y  = r1[0];
    b1v.x = r0[16]; b1v.y = r1[16];
    b2.x  = r0[32]; b2.y  = r1[32];
    b3.x  = r0[48]; b3.y  = r1[48];
    a0 = wmma_f32x4(a, b0,  a0);
    a1 = wmma_f32x4(a, b1v, a1);
    a2 = wmma_f32x4(a, b2,  a2);
    a3 = wmma_f32x4(a, b3,  a3);
  }

  float mean = estats[e * 4 + 0];
  float mnt  = estats[e * 4 + 1];
  float mxt  = estats[e * 4 + 2];
  float inv  = 1.0f / (mxt - mnt + EPS);

  int g = e >> 1;                          // E group 0..3
  size_t mbase = (e & 1) ? 16777216u : 0u; // M1 or M2
  int hi = (lane >> 4) * 8;
#pragma unroll
  for (int r = 0; r < 8; ++r) {
    int c = c0 + r + hi;
    const float* fr = f + (size_t)c * 16384 + p0 + l15;
    float* orow = out + mbase + ((size_t)bb * 256 + g * 64 + c) * 16384 + p0 + l15;
    float fv, tv;
    fv = fr[0];  tv = fmaxf(fv - mean, EPS); orow[0]  = (tv - mnt) * inv + fv + a0[r];
    fv = fr[16]; tv = fmaxf(fv - mean, EPS); orow[16] = (tv - mnt) * inv + fv + a1[r];
    fv = fr[32]; tv = fmaxf(fv - mean, EPS); orow[32] = (tv - mnt) * inv + fv + a2[r];
    fv = fr[48]; tv = fmaxf(fv - mean, EPS); orow[48] = (tv - mnt) * inv + fv + a3[r];
  }
}

// ---------------------------------------------------------------------------
// Kernel 8: nam, Ir/Vi activations and maps. One thread per (b,c,p) element.
// ---------------------------------------------------------------------------
__global__ __launch_bounds__(256) void k_maps(
    const float* __restrict__ cat, const float* __restrict__ am,
    const float* __restrict__ st, float* __restrict__ out)
{
  size_t i = (size_t)blockIdx.x * 256 + threadIdx.x;  // 4,194,304 total
  size_t bc = i >> 14;
  int p = (int)(i & 16383);
  int b = (int)(bc >> 6);
  float a  = am[b * 16384 + p];
  float mn = st[0], mx = st[1];
  float nam = (a - mn) / (mx - mn + EPS);
  float cf = cat[i];
  out[67108864u + i] = nam * cf;           // Ir_map (flat)
  out[71303168u + i] = (1.0f - nam) * cf;  // Vi_map (flat)
  if ((bc & 63) == 0) {
    out[75497472u + (size_t)b * 16384 + p] = nam;          // Ir_activation
    out[75563008u + (size_t)b * 16384 + p] = 1.0f - nam;   // Vi_activation
  }
}

// ---------------------------------------------------------------------------
// Host launcher
// ---------------------------------------------------------------------------
extern "C" void kernel_launch(void* const* d_in, const int* in_sizes, int n_in,
                              void* d_out, int out_size, void* d_ws, size_t ws_size,
                              hipStream_t stream) {
  (void)in_sizes; (void)n_in; (void)out_size; (void)ws_size;

  EPtrs ep;
  for (int i = 0; i < 8; ++i) ep.p[i] = (const float*)d_in[i];
  const float* fc  = (const float*)d_in[8];
  const float* w1  = (const float*)d_in[9];
  const float* b1  = (const float*)d_in[10];
  const float* w2  = (const float*)d_in[11];
  const float* b2  = (const float*)d_in[12];
  float* out = (float*)d_out;

  float* ws = (float*)d_ws;
  float* cat     = ws;                   // 16,777,216
  float* am      = ws + 16777216;        //     65,536
  float* scores  = ws + 16842752;        //    131,072
  float* psum    = ws + 16973824;        //      2,048
  float* pmin    = ws + 16975872;        //      2,048
  float* pmax    = ws + 16977920;        //      2,048
  float* estats  = ws + 16979968;        //         32
  float* namst   = ws + 16980000;        //          2
  float* spart   = ws + 16980032;        //  2,097,152
  float* wpack   = ws + 19077184;        //     73,728
  float* fcp     = ws + 19150912;        //  8,652,800 (4*128*130*130)

  // prep: repacked weights + zero-padded fc
  k_repack_w<<<288, 256, 0, stream>>>(w1, wpack);
  {
    dim3 g(67, 512);
    k_pad_fc<<<g, 256, 0, stream>>>(fc, fcp);
  }
  // conv1 + leaky relu (WMMA shifted GEMMs, branch-free K loop)
  k_conv1_wmma<<<4096, 32, 0, stream>>>(fcp, wpack, b1, cat);
  // conv2 -> activation maps
  k_conv2<<<256, 256, 0, stream>>>(cat, w2, b2, am);
  // norm_1 scalars of activation maps
  k_nam_minmax<<<1, 1024, 0, stream>>>(am, namst);
  // per-E global stats
  k_estats_part<<<2048, 256, 0, stream>>>(ep, psum, pmin, pmax);
  k_estats_final<<<8, 256, 0, stream>>>(psum, pmin, pmax, estats);
  // Gram (WMMA, K-split) + deterministic reduce
  k_gram_wmma<<<2048, 32, 0, stream>>>(ep, spart);
  k_gram_reduce<<<512, 256, 0, stream>>>(spart, scores);
  // column softmax over channel axis
  k_softmax<<<8, 256, 0, stream>>>(scores);
  // fw GEMM (WMMA) + fused intr epilogue -> M1/M2
  k_fw_wmma<<<32768, 32, 0, stream>>>(ep, scores, estats, out);
  // nam / Ir / Vi maps + activations
  k_maps<<<16384, 256, 0, stream>>>(cat, am, namst, out);
  // duplicated flattened outputs (M1 flat, M2 flat)
  hipMemcpyAsync(out + 33554432u, out,             (size_t)16777216 * 4,
                 hipMemcpyDeviceToDevice, stream);
  hipMemcpyAsync(out + 50331648u, out + 16777216u, (size_t)16777216 * 4,
                 hipMemcpyDeviceToDevice, stream);
}